// Attention_3D_52072183497198
// MI455X (gfx1250) — compile-verified
//
#include <hip/hip_runtime.h>
#include <math.h>

// ---------------- problem constants (from reference setup_inputs) -----------
constexpr int NB = 2;            // batch
constexpr int NC = 64;           // channels
constexpr int ND = 8, NH = 128, NW = 128;
constexpr int NN = ND * NH * NW; // 131072 spatial positions
constexpr int NCH3 = 3 * NC;     // 192 qkv channels
constexpr float FEPS = 1e-12f;

typedef __attribute__((ext_vector_type(2))) float v2f;
typedef __attribute__((ext_vector_type(8))) float v8f;

// D = A(16x4 f32) * B(4x16 f32) + C(16x16 f32)   — CDNA5 V_WMMA_F32_16X16X4_F32
static __device__ __forceinline__ v8f wmma_k4(v2f a, v2f b, v8f c) {
  return __builtin_amdgcn_wmma_f32_16x16x4_f32(false, a, false, b, (short)0, c,
                                               false, false);
}

// ---- CDNA5 async global->LDS copy (ASYNCcnt path, inline asm per ISA ch.10)
// LDS byte offset = low 32 bits of the generic pointer (flat->LDS mapping
// uses addr[31:0] per the aperture rules).
static __device__ __forceinline__ void async_ld_b128(float* lds_dst,
                                                     const float* gsrc) {
  unsigned loff = (unsigned)(uintptr_t)lds_dst;
  unsigned long long ga = (unsigned long long)(uintptr_t)gsrc;
  asm volatile("global_load_async_to_lds_b128 %0, %1, off"
               :: "v"(loff), "v"(ga) : "memory");
}
static __device__ __forceinline__ void wait_async0() {
  asm volatile("s_wait_asynccnt 0" ::: "memory");
}

// ---------------- kernel 0: zero the small accumulators ---------------------
__global__ void k_init(float* __restrict__ sq, float* __restrict__ raw) {
  int i = blockIdx.x * blockDim.x + threadIdx.x;
  if (i < NB * 128) sq[i] = 0.0f;
  if (i < NB * 4 * 256) raw[i] = 0.0f;
}

// ---------------- kernel 1: pointwise qkv conv (GEMM 192x64 @ 64xN) ---------
// one wave -> one 16(Mchan) x 16(Npos) f32 tile, K=64 via 16 WMMAs
__global__ void k_qkv(const float* __restrict__ x, const float* __restrict__ wqkv,
                      float* __restrict__ qkv) {
  const int NT = NN / 16;                               // 8192 spatial tiles
  int wid = blockIdx.x * (blockDim.x >> 5) + (threadIdx.x >> 5);
  int ot = wid % 12;                                    // 12 tiles of 16 out-chs
  int rest = wid / 12;
  int nt = rest & (NT - 1);
  int b = rest >> 13;
  int lane = threadIdx.x & 31;
  int m = lane & 15;                                    // row (A) / col (B,D)
  int kb = (lane >> 4) << 1;                            // K base: 0 or 2

  const float* Wr = wqkv + (size_t)(ot * 16 + m) * NC;  // A row
  const float* X = x + (size_t)b * NC * NN + (size_t)nt * 16 + m; // B col
  v8f c = {};
  for (int k0 = 0; k0 < NC; k0 += 4) {
    v2f a;  a.x = Wr[k0 + kb];            a.y = Wr[k0 + kb + 1];
    v2f bv; bv.x = X[(size_t)(k0 + kb) * NN]; bv.y = X[(size_t)(k0 + kb + 1) * NN];
    c = wmma_k4(a, bv, c);
  }
  int rowoff = (lane >> 4) * 8;
  float* O = qkv + (size_t)b * NCH3 * NN + (size_t)nt * 16 + m;
#pragma unroll
  for (int r = 0; r < 8; ++r)
    O[(size_t)(ot * 16 + rowoff + r) * NN] = c[r];
}

// ---------------- kernel 2: depthwise 3x3x3 conv + fused sum(q^2),sum(k^2) --
// Block = one (b,ch), 2 y-rows x 128 x of one z-plane. Halo tile (3 planes x
// 4 rows x 128) staged in LDS via async global->LDS b128 copies; OOB rows are
// zero-filled (implements SAME padding).
__global__ void k_dwconv(const float* __restrict__ qkv, const float* __restrict__ wdw,
                         float* __restrict__ dw, float* __restrict__ sq) {
  __shared__ __align__(16) float tile[3 * 4 * 128];     // [plane][row][x], 6 KB
  __shared__ float red[8];

  int blk = blockIdx.x;
  int nb = blk & (NN / 256 - 1); blk >>= 9;             // 512 chunks per channel
  int ch = blk % NCH3;
  int b = blk / NCH3;
  int nbase = nb * 256;                                 // never crosses a z-plane
  int z = nbase >> 14, y0 = (nbase >> 7) & 127;

  const float* src = qkv + ((size_t)b * NCH3 + ch) * NN;
  const float* kw = wdw + ch * 27;

  // ---- stage halo tile: 384 x b128 (16B) transfers, async to LDS ----
#pragma unroll
  for (int j = 0; j < 2; ++j) {
    int t4 = threadIdx.x + j * 256;
    if (t4 < 384) {
      int p = t4 >> 7;                                  // plane 0..2
      int r = (t4 >> 5) & 3;                            // row   0..3
      int xq = (t4 & 31) << 2;                          // x base (floats)
      int zz = z - 1 + p, yy = y0 - 1 + r;
      float* dst = &tile[t4 << 2];
      if (zz >= 0 && zz < ND && yy >= 0 && yy < NH) {
        async_ld_b128(dst, src + ((size_t)((zz << 7) + yy) << 7) + xq);
      } else {
        *(float4*)dst = make_float4(0.f, 0.f, 0.f, 0.f);
      }
    }
  }
  wait_async0();
  __syncthreads();

  // ---- compute one output element per thread from LDS ----
  int ylocal = threadIdx.x >> 7, x = threadIdx.x & 127;
  float acc = 0.0f;
#pragma unroll
  for (int dz = 0; dz < 3; ++dz) {
#pragma unroll
    for (int dy = 0; dy < 3; ++dy) {
      const float* row = &tile[dz * 512 + (ylocal + dy) * 128];
      float w0 = kw[dz * 9 + dy * 3 + 0];
      float w1 = kw[dz * 9 + dy * 3 + 1];
      float w2 = kw[dz * 9 + dy * 3 + 2];
      if (x > 0)   acc += w0 * row[x - 1];
      acc += w1 * row[x];
      if (x < 127) acc += w2 * row[x + 1];
    }
  }
  int n = nbase + threadIdx.x;
  dw[((size_t)b * NCH3 + ch) * NN + n] = acc;

  if (ch < 128) {                                       // q or k channel
    float s = acc * acc;
#pragma unroll
    for (int off = 16; off; off >>= 1) s += __shfl_xor(s, off, 32);
    int lane = threadIdx.x & 31, wv = threadIdx.x >> 5;
    if (lane == 0) red[wv] = s;
    __syncthreads();
    if (threadIdx.x == 0) {
      float t = 0.0f;
#pragma unroll
      for (int i = 0; i < 8; ++i) t += red[i];
      atomicAdd(&sq[b * 128 + ch], t);
    }
  }
}

// ---------------- kernel 3: raw q.k^T per head (16x16 tile = 2 heads) -------
// one wave accumulates a 512-long n-chunk with 128 WMMAs, atomic-adds tile
__global__ void k_attn(const float* __restrict__ dw, float* __restrict__ raw) {
  const int CHK = 512, NCHK = NN / CHK;                 // 256 chunks
  int wid = blockIdx.x * (blockDim.x >> 5) + (threadIdx.x >> 5);
  int chunk = wid % NCHK; wid /= NCHK;
  int hp = wid & 3;                                     // head pair
  int b = wid >> 2;
  int lane = threadIdx.x & 31;
  int m = lane & 15;
  int kb = (lane >> 4) << 1;

  const float* Q = dw + ((size_t)b * NCH3 + hp * 16 + m) * NN;       // A row m
  const float* K = dw + ((size_t)b * NCH3 + 64 + hp * 16 + m) * NN;  // B col m
  size_t n0 = (size_t)chunk * CHK;
  __builtin_prefetch(Q + n0, 0, 0);
  __builtin_prefetch(K + n0, 0, 0);
  v8f c = {};
  for (int k0 = 0; k0 < CHK; k0 += 4) {
    v2f a;  a.x = Q[n0 + k0 + kb];  a.y = Q[n0 + k0 + kb + 1];
    v2f bv; bv.x = K[n0 + k0 + kb]; bv.y = K[n0 + k0 + kb + 1];
    c = wmma_k4(a, bv, c);
  }
  float* R = raw + ((size_t)b * 4 + hp) * 256;
  int rowoff = (lane >> 4) * 8;
#pragma unroll
  for (int r = 0; r < 8; ++r)
    atomicAdd(&R[(rowoff + r) * 16 + m], c[r]);
}

// ---------------- kernel 4: normalize + softmax + fold W_proj ---------------
// M[b] = W_proj(64x64) * BlockDiag_h(softmax(q.k^T/(|q||k|) * T_h))
__global__ void k_fold(const float* __restrict__ raw, const float* __restrict__ sq,
                       const float* __restrict__ temp, const float* __restrict__ wproj,
                       float* __restrict__ Mm) {
  __shared__ float attn[64][8];                         // [h*8+c][e]
  int b = blockIdx.x;
  int tid = threadIdx.x;
  if (tid < 64) {
    int h = tid >> 3, cc = tid & 7;
    int hp = h >> 1, sub = (h & 1) * 8;
    const float* R = raw + ((size_t)b * 4 + hp) * 256 + (sub + cc) * 16 + sub;
    float qn = fmaxf(sqrtf(sq[b * 128 + h * 8 + cc]), FEPS);
    float t = temp[h];
    float vals[8], mx = -3.4e38f;
#pragma unroll
    for (int e = 0; e < 8; ++e) {
      float kn = fmaxf(sqrtf(sq[b * 128 + 64 + h * 8 + e]), FEPS);
      vals[e] = R[e] / (qn * kn) * t;
      mx = fmaxf(mx, vals[e]);
    }
    float sum = 0.0f;
#pragma unroll
    for (int e = 0; e < 8; ++e) { vals[e] = __expf(vals[e] - mx); sum += vals[e]; }
    float inv = 1.0f / sum;
#pragma unroll
    for (int e = 0; e < 8; ++e) attn[tid][e] = vals[e] * inv;
  }
  __syncthreads();
  for (int idx = tid; idx < 64 * 64; idx += blockDim.x) {
    int o = idx >> 6, eg = idx & 63, h = eg >> 3, e = eg & 7;
    float s = 0.0f;
#pragma unroll
    for (int cc = 0; cc < 8; ++cc)
      s += wproj[o * 64 + h * 8 + cc] * attn[h * 8 + cc][e];
    Mm[(size_t)b * 4096 + idx] = s;
  }
}

// ---------------- kernel 5: out = M[b](64x64) @ V(64xN) ---------------------
__global__ void k_outproj(const float* __restrict__ dw, const float* __restrict__ Mm,
                          float* __restrict__ out) {
  int wid = blockIdx.x * (blockDim.x >> 5) + (threadIdx.x >> 5);
  int nt = wid & 8191; wid >>= 13;
  int ot = wid & 3;
  int b = wid >> 2;
  int lane = threadIdx.x & 31;
  int m = lane & 15;
  int kb = (lane >> 4) << 1;

  const float* A = Mm + (size_t)b * 4096 + (size_t)(ot * 16 + m) * 64;
  const float* V = dw + ((size_t)b * NCH3 + 128) * NN + (size_t)nt * 16 + m;
  v8f c = {};
  for (int k0 = 0; k0 < 64; k0 += 4) {
    v2f a;  a.x = A[k0 + kb];                 a.y = A[k0 + kb + 1];
    v2f bv; bv.x = V[(size_t)(k0 + kb) * NN]; bv.y = V[(size_t)(k0 + kb + 1) * NN];
    c = wmma_k4(a, bv, c);
  }
  int rowoff = (lane >> 4) * 8;
  float* O = out + (size_t)b * NC * NN + (size_t)nt * 16 + m;
#pragma unroll
  for (int r = 0; r < 8; ++r)
    O[(size_t)(ot * 16 + rowoff + r) * NN] = c[r];
}

// ---------------- launcher ---------------------------------------------------
extern "C" void kernel_launch(void* const* d_in, const int* in_sizes, int n_in,
                              void* d_out, int out_size, void* d_ws, size_t ws_size,
                              hipStream_t stream) {
  const float* x     = (const float*)d_in[0];
  const float* wqkv  = (const float*)d_in[1];
  const float* wdw   = (const float*)d_in[2];
  const float* temp  = (const float*)d_in[3];
  const float* wproj = (const float*)d_in[4];
  float* out = (float*)d_out;

  float* ws  = (float*)d_ws;
  float* qkv = ws;                                      // [B,192,N]
  float* dwb = qkv + (size_t)NB * NCH3 * NN;            // [B,192,N] post-dwconv
  float* sq  = dwb + (size_t)NB * NCH3 * NN;            // [B,128] sum of squares
  float* raw = sq + NB * 128;                           // [B,4,16,16] raw q.k^T
  float* Mm  = raw + NB * 4 * 256;                      // [B,64,64] folded proj

  k_init<<<8, 256, 0, stream>>>(sq, raw);
  k_qkv<<<(NB * 12 * (NN / 16)) / 8, 256, 0, stream>>>(x, wqkv, qkv);
  k_dwconv<<<NB * NCH3 * (NN / 256), 256, 0, stream>>>(qkv, wdw, dwb, sq);
  k_attn<<<(NB * 4 * 256) / 8, 256, 0, stream>>>(dwb, raw);
  k_fold<<<NB, 256, 0, stream>>>(raw, sq, temp, wproj, Mm);
  k_outproj<<<(NB * 4 * 8192) / 8, 256, 0, stream>>>(dwb, Mm, out);
}